// HeterogeneousGNN_82068235092029
// MI455X (gfx1250) — compile-verified
//
#include <hip/hip_runtime.h>

typedef __attribute__((ext_vector_type(2))) float v2f;
typedef __attribute__((ext_vector_type(8))) float v8f;

#define WMMA_F32_4(a, b, c) \
  __builtin_amdgcn_wmma_f32_16x16x4_f32(false, (a), false, (b), (short)0, (c), false, false)

// ---------------------------------------------------------------------------
// Generic [rows x FIN] @ [FIN x 64] + bias encoder, fp32 WMMA 16x16x4.
// Block = 128 threads = 4 waves; wave w computes output columns [16w, 16w+16).
// Grid.x = rows/16 (rows is a multiple of 16).
// ---------------------------------------------------------------------------
template <int FIN>
__global__ __launch_bounds__(128) void encode_wmma(
    const float* __restrict__ in, const float* __restrict__ W,
    const float* __restrict__ bias, float* __restrict__ xout, int row_off) {
  const int lane = threadIdx.x & 31;
  const int wave = threadIdx.x >> 5;
  const int n0 = wave * 16;
  const int m = lane & 15;   // A row within tile / B,D column within tile
  const int kh = lane >> 4;  // K-half select for A/B, M-half select for D
  const int row = blockIdx.x * 16 + m;
  const float* __restrict__ arow = in + (size_t)row * FIN;

  v8f c = {};
#pragma unroll
  for (int k0 = 0; k0 < FIN; k0 += 4) {
    const int k = k0 + 2 * kh;
    v2f a;
    a.x = arow[k];
    a.y = arow[k + 1];
    v2f b;
    b.x = W[(size_t)k * 64 + n0 + m];
    b.y = W[(size_t)(k + 1) * 64 + n0 + m];
    c = WMMA_F32_4(a, b, c);
  }
  const float bb = bias[n0 + m];
  float* __restrict__ orow = xout + (size_t)(row_off + blockIdx.x * 16) * 64;
#pragma unroll
  for (int v = 0; v < 8; ++v) {
    const int M = v + 8 * kh;
    orow[(size_t)M * 64 + n0 + m] = c[v] + bb;
  }
}

// ---------------------------------------------------------------------------
// Fused SAGE dense step: out = relu((agg*inv) @ Wl + x @ Wr + b), H=64.
// Same tiling as encoder; two WMMAs per K-step into one accumulator.
// ---------------------------------------------------------------------------
__global__ __launch_bounds__(128) void sage_wmma(
    const float* __restrict__ agg, const float* __restrict__ inv,
    const float* __restrict__ xin, const float* __restrict__ Wl,
    const float* __restrict__ Wr, const float* __restrict__ bias,
    float* __restrict__ xout) {
  const int lane = threadIdx.x & 31;
  const int wave = threadIdx.x >> 5;
  const int n0 = wave * 16;
  const int m = lane & 15;
  const int kh = lane >> 4;
  const int row = blockIdx.x * 16 + m;
  const float invm = inv[row];
  const float* __restrict__ ga = agg + (size_t)row * 64;
  const float* __restrict__ gx = xin + (size_t)row * 64;

  v8f c = {};
#pragma unroll
  for (int k0 = 0; k0 < 64; k0 += 4) {
    const int k = k0 + 2 * kh;
    v2f am;
    am.x = ga[k] * invm;
    am.y = ga[k + 1] * invm;
    v2f bl;
    bl.x = Wl[(size_t)k * 64 + n0 + m];
    bl.y = Wl[(size_t)(k + 1) * 64 + n0 + m];
    c = WMMA_F32_4(am, bl, c);
    v2f ax;
    ax.x = gx[k];
    ax.y = gx[k + 1];
    v2f br;
    br.x = Wr[(size_t)k * 64 + n0 + m];
    br.y = Wr[(size_t)(k + 1) * 64 + n0 + m];
    c = WMMA_F32_4(ax, br, c);
  }
  const float bb = bias[n0 + m];
  float* __restrict__ orow = xout + (size_t)(blockIdx.x * 16) * 64;
#pragma unroll
  for (int v = 0; v < 8; ++v) {
    const int M = v + 8 * kh;
    const float val = c[v] + bb;
    orow[(size_t)M * 64 + n0 + m] = val > 0.0f ? val : 0.0f;
  }
}

// ---------------------------------------------------------------------------
// Aggregation helpers
// ---------------------------------------------------------------------------
__global__ __launch_bounds__(256) void zero_f32(float* __restrict__ p, long long n) {
  const long long i = (long long)blockIdx.x * blockDim.x + threadIdx.x;
  if (i < n) p[i] = 0.0f;
}

__global__ __launch_bounds__(256) void count_edges(const int* __restrict__ dst,
                                                   float* __restrict__ cnt, int E) {
  const int e = blockIdx.x * blockDim.x + threadIdx.x;
  if (e < E) atomicAdd(cnt + dst[e], 1.0f);
}

__global__ __launch_bounds__(256) void make_inv(float* __restrict__ c, int N) {
  const int i = blockIdx.x * blockDim.x + threadIdx.x;
  if (i < N) c[i] = 1.0f / fmaxf(c[i], 1.0f);
}

// 16 threads per edge, 4 floats each: float4 gather from x[src], 4x f32 atomic
// adds into agg[dst]. x and agg are L2-resident (76.8 MB each, 192 MB L2).
__global__ __launch_bounds__(256) void scatter_add(
    const int* __restrict__ src, const int* __restrict__ dst,
    const float* __restrict__ x, float* __restrict__ agg, int E) {
  const long long t = (long long)blockIdx.x * blockDim.x + threadIdx.x;
  const int e = (int)(t >> 4);
  if (e >= E) return;
  const int f = ((int)t & 15) * 4;
  const float4 v = *reinterpret_cast<const float4*>(x + (size_t)src[e] * 64 + f);
  float* ap = agg + (size_t)dst[e] * 64 + f;
  atomicAdd(ap + 0, v.x);
  atomicAdd(ap + 1, v.y);
  atomicAdd(ap + 2, v.z);
  atomicAdd(ap + 3, v.w);
}

// ---------------------------------------------------------------------------
// Fused classifier: out = relu(x@Wc1 + bc1) @ Wc2 + bc2  (64 -> 32 -> 2)
// Weight reads are wave-uniform -> scalar loads; x row lives in VGPRs.
// ---------------------------------------------------------------------------
__global__ __launch_bounds__(256) void classifier(
    const float* __restrict__ x, const float* __restrict__ Wc1,
    const float* __restrict__ bc1, const float* __restrict__ Wc2,
    const float* __restrict__ bc2, float* __restrict__ out, int N) {
  const int i = blockIdx.x * blockDim.x + threadIdx.x;
  if (i >= N) return;
  float xr[64];
  const float* __restrict__ row = x + (size_t)i * 64;
#pragma unroll
  for (int k = 0; k < 64; ++k) xr[k] = row[k];
  float o0 = bc2[0], o1 = bc2[1];
#pragma unroll 4
  for (int j = 0; j < 32; ++j) {
    float h = bc1[j];
#pragma unroll
    for (int k = 0; k < 64; ++k) h = fmaf(xr[k], Wc1[k * 32 + j], h);
    h = h > 0.0f ? h : 0.0f;
    o0 = fmaf(h, Wc2[j * 2 + 0], o0);
    o1 = fmaf(h, Wc2[j * 2 + 1], o1);
  }
  out[(size_t)i * 2 + 0] = o0;
  out[(size_t)i * 2 + 1] = o1;
}

// ---------------------------------------------------------------------------
extern "C" void kernel_launch(void* const* d_in, const int* in_sizes, int n_in,
                              void* d_out, int out_size, void* d_ws, size_t ws_size,
                              hipStream_t stream) {
  constexpr int N_IND = 100000, N_COM = 100000, N_TRU = 100000;
  constexpr int N = N_IND + N_COM + N_TRU;  // 300000, multiple of 16
  const int E = in_sizes[3] / 2;            // 4,800,000

  const float* x_ind = (const float*)d_in[0];
  const float* x_com = (const float*)d_in[1];
  const float* x_tru = (const float*)d_in[2];
  const int* ei = (const int*)d_in[3];  // [2, E] int32: src row then dst row
  const float* W_ind = (const float*)d_in[4];
  const float* b_ind = (const float*)d_in[5];
  const float* W_com = (const float*)d_in[6];
  const float* b_com = (const float*)d_in[7];
  const float* W_tru = (const float*)d_in[8];
  const float* b_tru = (const float*)d_in[9];
  const float* W1l = (const float*)d_in[10];
  const float* W1r = (const float*)d_in[11];
  const float* b1 = (const float*)d_in[12];
  const float* W2l = (const float*)d_in[13];
  const float* W2r = (const float*)d_in[14];
  const float* b2 = (const float*)d_in[15];
  const float* Wc1 = (const float*)d_in[16];
  const float* bc1 = (const float*)d_in[17];
  const float* Wc2 = (const float*)d_in[18];
  const float* bc2 = (const float*)d_in[19];

  const int* src = ei;
  const int* dst = ei + E;

  // Workspace layout (floats): x0[N*64] | agg[N*64] | x1[N*64] | cnt[N]
  float* ws = (float*)d_ws;
  float* x0 = ws;
  float* agg = x0 + (size_t)N * 64;
  float* x1 = agg + (size_t)N * 64;
  float* cnt = x1 + (size_t)N * 64;

  const long long NH = (long long)N * 64;

  // 1) Per-type encoders -> x0
  encode_wmma<32><<<N_IND / 16, 128, 0, stream>>>(x_ind, W_ind, b_ind, x0, 0);
  encode_wmma<48><<<N_COM / 16, 128, 0, stream>>>(x_com, W_com, b_com, x0, N_IND);
  encode_wmma<24><<<N_TRU / 16, 128, 0, stream>>>(x_tru, W_tru, b_tru, x0, N_IND + N_COM);

  // 2) Degrees -> inverse mean denominators (shared by both layers)
  zero_f32<<<(N + 255) / 256, 256, 0, stream>>>(cnt, N);
  count_edges<<<(E + 255) / 256, 256, 0, stream>>>(dst, cnt, E);
  make_inv<<<(N + 255) / 256, 256, 0, stream>>>(cnt, N);

  const long long scatterThreads = (long long)E * 16;
  const int scatterBlocks = (int)((scatterThreads + 255) / 256);

  // 3) SAGE layer 1: agg = scatter(x0), x1 = relu(mean@W1l + x0@W1r + b1)
  zero_f32<<<(int)((NH + 255) / 256), 256, 0, stream>>>(agg, NH);
  scatter_add<<<scatterBlocks, 256, 0, stream>>>(src, dst, x0, agg, E);
  sage_wmma<<<N / 16, 128, 0, stream>>>(agg, cnt, x0, W1l, W1r, b1, x1);

  // 4) SAGE layer 2: agg = scatter(x1), x0 = relu(mean@W2l + x1@W2r + b2)
  zero_f32<<<(int)((NH + 255) / 256), 256, 0, stream>>>(agg, NH);
  scatter_add<<<scatterBlocks, 256, 0, stream>>>(src, dst, x1, agg, E);
  sage_wmma<<<N / 16, 128, 0, stream>>>(agg, cnt, x1, W2l, W2r, b2, x0);

  // 5) Classifier MLP -> d_out [N, 2]
  classifier<<<(N + 255) / 256, 256, 0, stream>>>(x0, Wc1, bc1, Wc2, bc2,
                                                  (float*)d_out, N);
}